// LSEPLoss_65670049955848
// MI455X (gfx1250) — compile-verified
//
#include <hip/hip_runtime.h>

typedef float v2f __attribute__((ext_vector_type(2)));
typedef float v4f __attribute__((ext_vector_type(4)));
typedef float v8f __attribute__((ext_vector_type(8)));
typedef int   v4i __attribute__((ext_vector_type(4)));

#define BATCH   32768
#define NCLS    1000
#define ROWS_PER_BLOCK 16
#define NBLOCKS (BATCH / ROWS_PER_BLOCK)   // 2048 blocks, 16384 waves

// ---------------------------------------------------------------------------
// Kernel 1: per-block (16-row tile) partial  sum_m s_neg[m]*s_pos[m]
//   - 8 waves per block split the 1000 columns: waves 0..6 -> 128 cols,
//     wave 7 -> 104 cols (all multiples of 8 -> chunks of 8 cols, b128 loads)
//   - row sums on the matrix pipe: v_wmma_f32_16x16x4_f32 with B = ones
//     (column->K assignment is arbitrary for a full row sum, so contiguous
//      float4 per lane needs no cross-lane shuffles)
//   - wave index forced to an SGPR via readfirstlane so the column loop is
//     scalar/uniform: no exec-mask loop, EXEC provably all-ones at the WMMAs,
//     and the unroll can actually fire (more b128 loads in flight)
// ---------------------------------------------------------------------------
__global__ __launch_bounds__(256) void lsep_partial_kernel(
    const float* __restrict__ in,
    const int*   __restrict__ tgt,
    float*       __restrict__ partial)
{
    __shared__ float s_neg[8][16];
    __shared__ float s_pos[8][16];
    __shared__ float s_prod[16];

    const int tid  = threadIdx.x;
    const int wave = __builtin_amdgcn_readfirstlane(tid >> 5);  // SGPR, 0..7
    const int lane = tid & 31;
    const int m    = lane & 15;     // row within 16-row tile (WMMA M index)
    const int h    = lane >> 4;     // half: K-group / column sub-offset

    const long rowBase  = (long)(blockIdx.x * ROWS_PER_BLOCK + m) * NCLS;
    const int  colStart = wave * 128;
    const int  nChunks  = (wave == 7) ? 13 : 16;   // scalar trip count

    v8f cneg = {};                 // 16x16 f32 accumulator (row sums, cols identical)
    v8f cpos = {};
    const v2f ones = {1.0f, 1.0f}; // B = all ones -> D[m][*] = sum_k A[m][k]

    const float* pIn = in  + rowBase + colStart + 4 * h;
    const int*   pT  = tgt + rowBase + colStart + 4 * h;

    #pragma unroll 2
    for (int c = 0; c < nChunks; ++c) {
        // b128 streaming loads, non-temporal (touch-once, > L2 footprint)
        const v4f x = __builtin_nontemporal_load((const v4f*)(pIn + 8 * c));
        const v4i t = __builtin_nontemporal_load((const v4i*)(pT  + 8 * c));

        // one exp per element: exp(x) if t==0 (neg), exp(-x) if t!=0 (pos)
        const float e0 = __expf(t.x ? -x.x : x.x);
        const float e1 = __expf(t.y ? -x.y : x.y);
        const float e2 = __expf(t.z ? -x.z : x.z);
        const float e3 = __expf(t.w ? -x.w : x.w);

        const v2f an0 = { t.x ? 0.0f : e0, t.y ? 0.0f : e1 };
        const v2f an1 = { t.z ? 0.0f : e2, t.w ? 0.0f : e3 };
        const v2f ap0 = { t.x ? e0 : 0.0f, t.y ? e1 : 0.0f };
        const v2f ap1 = { t.z ? e2 : 0.0f, t.w ? e3 : 0.0f };

        // D = A x ones + C : accumulates 16 row-sums per call, full f32
        cneg = __builtin_amdgcn_wmma_f32_16x16x4_f32(false, an0, false, ones,
                                                     (short)0, cneg, false, false);
        cneg = __builtin_amdgcn_wmma_f32_16x16x4_f32(false, an1, false, ones,
                                                     (short)0, cneg, false, false);
        cpos = __builtin_amdgcn_wmma_f32_16x16x4_f32(false, ap0, false, ones,
                                                     (short)0, cpos, false, false);
        cpos = __builtin_amdgcn_wmma_f32_16x16x4_f32(false, ap1, false, ones,
                                                     (short)0, cpos, false, false);
    }

    // C/D layout: VGPR j holds row (j + 8*half) replicated across its 16
    // lanes. Lanes 0 and 16 together carry all 16 partial row sums.
    if (lane == 0 || lane == 16) {
        const int rbase = (lane == 0) ? 0 : 8;
        #pragma unroll
        for (int j = 0; j < 8; ++j) {
            s_neg[wave][rbase + j] = cneg[j];
            s_pos[wave][rbase + j] = cpos[j];
        }
    }
    __syncthreads();

    // combine the 8 column-segments per row (fixed order -> deterministic),
    // THEN take the product, then sum the 16 rows
    if (tid < 16) {
        float sn = 0.0f, sp = 0.0f;
        #pragma unroll
        for (int w = 0; w < 8; ++w) { sn += s_neg[w][tid]; sp += s_pos[w][tid]; }
        s_prod[tid] = sn * sp;
    }
    __syncthreads();

    if (tid == 0) {
        float tot = 0.0f;
        #pragma unroll
        for (int r = 0; r < 16; ++r) tot += s_prod[r];
        partial[blockIdx.x] = tot;
    }
}

// ---------------------------------------------------------------------------
// Kernel 2: deterministic fixed-order reduction of 2048 block partials in
// double, then log1p. Single 256-thread block; negligible runtime.
// ---------------------------------------------------------------------------
__global__ __launch_bounds__(256) void lsep_final_kernel(
    const float* __restrict__ partial,
    float*       __restrict__ out)
{
    __shared__ double sd[256];
    const int tid = threadIdx.x;

    double acc = 0.0;
    for (int i = tid; i < NBLOCKS; i += 256)
        acc += (double)partial[i];
    sd[tid] = acc;
    __syncthreads();

    #pragma unroll
    for (int s = 128; s > 0; s >>= 1) {
        if (tid < s) sd[tid] += sd[tid + s];
        __syncthreads();
    }
    if (tid == 0)
        out[0] = (float)log1p(sd[0]);
}

extern "C" void kernel_launch(void* const* d_in, const int* in_sizes, int n_in,
                              void* d_out, int out_size, void* d_ws, size_t ws_size,
                              hipStream_t stream)
{
    const float* in  = (const float*)d_in[0];
    const int*   tgt = (const int*)d_in[1];
    float* partial = (float*)d_ws;          // 2048 floats = 8 KB scratch
    float* out     = (float*)d_out;

    lsep_partial_kernel<<<NBLOCKS, 256, 0, stream>>>(in, tgt, partial);
    lsep_final_kernel<<<1, 256, 0, stream>>>(partial, out);
}